// IsometryReg_67843303408032
// MI455X (gfx1250) — compile-verified
//
#include <hip/hip_runtime.h>
#include <math.h>

typedef float v2f __attribute__((ext_vector_type(2)));
typedef float v8f __attribute__((ext_vector_type(8)));

#define Bn 2048
#define Dn 3072
#define Cn 10
#define Mn 9
#define NTILE (Bn / 16)        // 128 tiles
#define NFRAG (Dn / 4)         // 768 B-fragments

// scratch layout (in floats)
#define WS_G    0              // 100 floats: G = W^T W
#define WS_PART 128            // 128 floats: per-tile partial sums
#define WS_WP   256            // 768*32 v2f = 49152 floats: packed B fragments

__constant__ float kNumStab = 1e-7f;

// ---------------- Kernel 1: G = W^T W (10x10), fully deterministic ----------------
__global__ void k_gram(const float* __restrict__ W, float* ws) {
    int t = threadIdx.x;
    if (t < Cn * Cn) {
        int k = t / Cn, l = t % Cn;
        float acc = 0.f;
        for (int d = 0; d < Dn; ++d)
            acc = fmaf(W[d * Cn + k], W[d * Cn + l], acc);
        ws[WS_G + t] = acc;
    }
}

// ---------------- Kernel 2: pack W into WMMA B-fragment order ----------------
// B (4x16 f32) layout: VGPR j, lane-half h, col n=lane&15 holds K = kk + 2h + j.
__global__ void k_wprep(const float* __restrict__ W, float* ws) {
    int gid  = blockIdx.x * blockDim.x + threadIdx.x;   // 0 .. 768*32-1
    int frag = gid >> 5;
    int lane = gid & 31;
    int h = lane >> 4, n = lane & 15;
    int kk = frag * 4 + 2 * h;
    float x = 0.f, y = 0.f;
    if (n < Cn) {
        x = W[kk * Cn + n];
        y = W[(kk + 1) * Cn + n];
    }
    float2 v = make_float2(x, y);
    reinterpret_cast<float2*>(ws + WS_WP)[gid] = v;
}

// ---------------- Kernel 3: logits GEMM via V_WMMA_F32_16X16X4_F32 + epilogue ----------------
__launch_bounds__(128)
__global__ void k_main(const float* __restrict__ data,
                       const float* __restrict__ bias,
                       float* ws) {
    __shared__ float cf[4][8][32];   // per-wave C fragments
    __shared__ float Gs[Cn * Cn];
    __shared__ float red[16];

    const int tid  = threadIdx.x;
    const int wave = tid >> 5;
    const int lane = tid & 31;
    if (tid < Cn * Cn) Gs[tid] = ws[WS_G + tid];

    const int tile = blockIdx.x;
    const int row0 = tile * 16;
    const int h    = lane >> 4;      // lane half selects K pair
    const int mrow = lane & 15;      // A-matrix row (sample within tile)

    // each wave covers 768 of the 3072 K values
    const int kbeg = wave * (Dn / 4);
    const int kend = kbeg + (Dn / 4);

    const float* pA = data + (size_t)(row0 + mrow) * Dn + kbeg + 2 * h;
    const v2f*   pB = reinterpret_cast<const v2f*>(ws + WS_WP) + (kbeg >> 2) * 32 + lane;

    v8f c = {0.f, 0.f, 0.f, 0.f, 0.f, 0.f, 0.f, 0.f};
    #pragma unroll 8
    for (int kk = kbeg; kk < kend; kk += 4) {
        v2f a = *reinterpret_cast<const v2f*>(pA);
        v2f b = *pB;
        // D = A(16x4 f32) * B(4x16 f32) + C, exact fp32
        c = __builtin_amdgcn_wmma_f32_16x16x4_f32(false, a, false, b,
                                                  (short)0, c, false, false);
        pA += 4;
        pB += 32;
    }

    #pragma unroll
    for (int r = 0; r < 8; ++r) cf[wave][r][lane] = c[r];
    __syncthreads();

    if (tid < 16) {
        // C layout: VGPR r, lane-half hh -> sample M = r + 8*hh, class N = lane&15
        const int s  = tid;
        const int hh = s >> 3, rr_ = s & 7;

        float z[Cn];
        #pragma unroll
        for (int k = 0; k < Cn; ++k)
            z[k] = cf[0][rr_][hh * 16 + k] + cf[1][rr_][hh * 16 + k] +
                   cf[2][rr_][hh * 16 + k] + cf[3][rr_][hh * 16 + k] + bias[k];

        // softmax -> probs -> sqrt
        float mx = z[0];
        #pragma unroll
        for (int k = 1; k < Cn; ++k) mx = fmaxf(mx, z[k]);
        float e[Cn]; float S = 0.f;
        #pragma unroll
        for (int k = 0; k < Cn; ++k) { e[k] = expf(z[k] - mx); S += e[k]; }

        const float ns = kNumStab;
        const float ac = 1.0f - (float)Cn * ns;
        float sm[Cn], nc[Cn], al[Cn];
        float invS = 1.0f / S, ncsum = 0.f;
        #pragma unroll
        for (int k = 0; k < Cn; ++k) {
            sm[k] = e[k] * invS;
            float p = ac * sm[k] + ns;
            nc[k] = sqrtf(p);
            al[k] = ac * sm[k] / (2.0f * nc[k]);   // d(sqrt p)/dlogit coefficient
            ncsum += nc[k];
        }
        const float u = 1.0f - nc[Cn - 1];

        // A_ik = c1[i]*d_ik + c2[i]*d_9k - (c1[i]+c2[i])*s_k
        float c1[Mn], c2[Mn];
        #pragma unroll
        for (int i = 0; i < Mn; ++i) {
            c1[i] = 2.0f * al[i] / u;
            c2[i] = 2.0f * nc[i] * al[Cn - 1] / (u * u);
        }

        // q = G s (G symmetric), r = q.s
        float q[Cn]; float rr = 0.f;
        #pragma unroll
        for (int l = 0; l < Cn; ++l) {
            float acc = 0.f;
            #pragma unroll
            for (int k = 0; k < Cn; ++k) acc = fmaf(Gs[k * Cn + l], sm[k], acc);
            q[l] = acc;
            rr = fmaf(acc, sm[l], rr);
        }

        const float rsqrt10 = 0.31622776601683794f;
        float delta  = 2.0f * acosf(fminf(1.0f, ncsum * rsqrt10));
        float factor = delta * delta / (4.0f * u * u * 0.01f);   // eps^2 = 0.01

        // ||A G A^T - factor I||_F^2 via closed form
        float fsum = 0.f;
        #pragma unroll
        for (int i = 0; i < Mn; ++i) {
            float ci  = c1[i] + c2[i];
            float Ti9 = c1[i] * Gs[i * Cn + (Cn - 1)] + c2[i] * Gs[(Cn - 1) * Cn + (Cn - 1)] - ci * q[Cn - 1];
            float Tis = c1[i] * q[i] + c2[i] * q[Cn - 1] - ci * rr;
            #pragma unroll
            for (int j = 0; j < Mn; ++j) {
                float Tij = c1[i] * Gs[i * Cn + j] + c2[i] * Gs[(Cn - 1) * Cn + j] - ci * q[j];
                float g   = c1[j] * Tij + c2[j] * Ti9 - (c1[j] + c2[j]) * Tis;
                float d   = g - (i == j ? factor : 0.0f);
                fsum = fmaf(d, d, fsum);
            }
        }
        red[tid] = sqrtf(fsum);
    }
    __syncthreads();

    if (tid == 0) {
        float t = 0.f;
        #pragma unroll
        for (int i = 0; i < 16; ++i) t += red[i];
        ws[WS_PART + tile] = t;
    }
}

// ---------------- Kernel 4: deterministic final reduction ----------------
__global__ void k_final(const float* __restrict__ ws, float* __restrict__ out) {
    __shared__ float red[NTILE];
    int t = threadIdx.x;
    red[t] = ws[WS_PART + t];
    __syncthreads();
    for (int off = NTILE / 2; off > 0; off >>= 1) {
        if (t < off) red[t] += red[t + off];
        __syncthreads();
    }
    if (t == 0) out[0] = red[0] / ((float)Bn * (float)(Mn * Mn));
}

extern "C" void kernel_launch(void* const* d_in, const int* in_sizes, int n_in,
                              void* d_out, int out_size, void* d_ws, size_t ws_size,
                              hipStream_t stream) {
    const float* data = (const float*)d_in[0];   // [2048,3,32,32] f32
    const float* W    = (const float*)d_in[1];   // [3072,10] f32
    const float* b    = (const float*)d_in[2];   // [10] f32
    float* ws  = (float*)d_ws;
    float* out = (float*)d_out;

    k_gram <<<1, 128, 0, stream>>>(W, ws);
    k_wprep<<<(NFRAG * 32) / 256, 256, 0, stream>>>(W, ws);
    k_main <<<NTILE, 128, 0, stream>>>(data, b, ws);
    k_final<<<1, NTILE, 0, stream>>>(ws, out);
}